// BoundaryLoss_88038239634235
// MI455X (gfx1250) — compile-verified
//
#include <hip/hip_runtime.h>
#include <stdint.h>

#define XDIM 192
#define YDIM 192
#define ZDIM 96
#define BDIM 4
#define WRDX 6                                   // 32-bit words along X
#define NWORDS (BDIM * ZDIM * YDIM * WRDX)       // 442,368 words per mask
#define VOL (ZDIM * YDIM * XDIM)                 // 3,538,944 voxels per batch
#define TOTV ((long)BDIM * VOL)                  // 14,155,776 voxels
#define ITERS 20

typedef float v2f __attribute__((ext_vector_type(2)));
typedef float v8f __attribute__((ext_vector_type(8)));

#if __has_builtin(__builtin_amdgcn_global_load_async_to_lds_b32)
#define HAVE_ASYNC_LDS 1
#else
#define HAVE_ASYNC_LDS 0
#endif

typedef __attribute__((address_space(1))) int* gptr_t;
typedef __attribute__((address_space(3))) int* lptr_t;

// ---------------------------------------------------------------- init
__global__ __launch_bounds__(32) void bl_init(int* fg, float* out) {
    if (threadIdx.x < 4) fg[threadIdx.x] = 0;
    if (threadIdx.x == 0) out[0] = 0.0f;
}

// ---------------------------------------------------------------- pack
// One thread per 32-bit mask word: pack 32 int32 targets into bits,
// init erosion (E) and dilation (D) working masks, zero byte counters,
// wave-reduce popcount -> per-batch fg count.
__global__ __launch_bounds__(256) void bl_pack(const int* __restrict__ tgt,
                                               uint32_t* __restrict__ mask0,
                                               uint32_t* __restrict__ E0,
                                               uint32_t* __restrict__ D0,
                                               uint32_t* __restrict__ cntIn,
                                               uint32_t* __restrict__ cntOut,
                                               int* __restrict__ fg) {
    const int w = blockIdx.x * blockDim.x + threadIdx.x;  // < NWORDS
    const long v0 = (long)w * 32;
    uint32_t bits = 0u;
#pragma unroll
    for (int i = 0; i < 32; ++i)
        bits |= (uint32_t)(tgt[v0 + i] & 1) << i;
    mask0[w] = bits;
    E0[w] = bits;
    D0[w] = bits;
    const uint4 z = {0u, 0u, 0u, 0u};
    ((uint4*)cntIn)[(size_t)w * 2 + 0] = z;
    ((uint4*)cntIn)[(size_t)w * 2 + 1] = z;
    ((uint4*)cntOut)[(size_t)w * 2 + 0] = z;
    ((uint4*)cntOut)[(size_t)w * 2 + 1] = z;
    int pc = __popc(bits);
#pragma unroll
    for (int off = 16; off; off >>= 1) pc += __shfl_xor(pc, off, 32);
    if ((threadIdx.x & 31) == 0) {
        const int b = w / (ZDIM * YDIM * WRDX);  // 110,592 words/batch, wave-aligned
        atomicAdd(&fg[b], pc);
    }
}

// spread low nibble bits to byte lanes of a dword (byte-counter add mask)
__device__ __forceinline__ uint32_t spread4(uint32_t x) {
    return (x & 1u) | ((x & 2u) << 7) | ((x & 4u) << 14) | ((x & 8u) << 21);
}

// ---------------------------------------------------------------- erosion step
// Tile: full X row (6 words) x 16 Y x 8 Z output, halo 1 in Y/Z staged in LDS
// via global_load_async_to_lds_b32 (ASYNCcnt-tracked).
// E <- erode(E)  (27-neighborhood AND; out-of-volume acts as 1s, matching inf-pad min)
// D <- dilate(D) (27-neighborhood OR;  out-of-volume acts as 0s)
// cnt_in  += E bit per voxel   (accumulated erosion of t)
// cnt_out += ~D bit per voxel  (accumulated erosion of 1-t)
__global__ __launch_bounds__(256) void bl_erode(const uint32_t* __restrict__ Ein,
                                                const uint32_t* __restrict__ Din,
                                                uint32_t* __restrict__ Eout,
                                                uint32_t* __restrict__ Dout,
                                                uint32_t* __restrict__ cntIn,
                                                uint32_t* __restrict__ cntOut) {
    __shared__ uint32_t tE[10 * 18 * WRDX];
    __shared__ uint32_t tD[10 * 18 * WRDX];
    const int tid = threadIdx.x;
    const int ty = blockIdx.x % (YDIM / 16);
    const int tz = (blockIdx.x / (YDIM / 16)) % (ZDIM / 8);
    const int bb = blockIdx.x / ((YDIM / 16) * (ZDIM / 8));
    const int y0 = ty * 16, z0 = tz * 8;

    for (int i = tid; i < 10 * 18 * WRDX; i += 256) {
        const int wx = i % WRDX;
        const int hy = (i / WRDX) % 18;
        const int hz = i / (WRDX * 18);
        const int gy = y0 + hy - 1;
        const int gz = z0 + hz - 1;
        if (gy >= 0 && gy < YDIM && gz >= 0 && gz < ZDIM) {
            const int gw = ((bb * ZDIM + gz) * YDIM + gy) * WRDX + wx;
#if HAVE_ASYNC_LDS
            __builtin_amdgcn_global_load_async_to_lds_b32(
                (gptr_t)(Ein + gw), (lptr_t)(&tE[i]), 0, 0);
            __builtin_amdgcn_global_load_async_to_lds_b32(
                (gptr_t)(Din + gw), (lptr_t)(&tD[i]), 0, 0);
#else
            tE[i] = Ein[gw];
            tD[i] = Din[gw];
#endif
        } else {
            tE[i] = 0xFFFFFFFFu;  // erosion: OOB never erodes
            tD[i] = 0u;           // dilation: OOB never dilates
        }
    }
#if HAVE_ASYNC_LDS
#if __has_builtin(__builtin_amdgcn_s_wait_asynccnt)
    __builtin_amdgcn_s_wait_asynccnt(0);
#else
    asm volatile("s_wait_asynccnt 0" ::: "memory");
#endif
#endif
    __syncthreads();

    for (int o = tid; o < WRDX * 16 * 8; o += 256) {
        const int wx = o % WRDX;
        const int ly = (o / WRDX) % 16;
        const int lz = o / (WRDX * 16);
        uint32_t e = 0xFFFFFFFFu;
        uint32_t d = 0u;
#pragma unroll
        for (int dz = 0; dz < 3; ++dz) {
#pragma unroll
            for (int dy = 0; dy < 3; ++dy) {
                const uint32_t* rE = &tE[((lz + dz) * 18 + (ly + dy)) * WRDX];
                const uint32_t* rD = &tD[((lz + dz) * 18 + (ly + dy)) * WRDX];
                const uint32_t w = rE[wx];
                const uint32_t wl = (wx > 0) ? rE[wx - 1] : 0xFFFFFFFFu;
                const uint32_t wr = (wx < WRDX - 1) ? rE[wx + 1] : 0xFFFFFFFFu;
                e &= w & ((w << 1) | (wl >> 31)) & ((w >> 1) | (wr << 31));
                const uint32_t v = rD[wx];
                const uint32_t vl = (wx > 0) ? rD[wx - 1] : 0u;
                const uint32_t vr = (wx < WRDX - 1) ? rD[wx + 1] : 0u;
                d |= v | (v << 1) | (vl >> 31) | (v >> 1) | (vr << 31);
            }
        }
        const int gy = y0 + ly, gz = z0 + lz;
        const int gw = ((bb * ZDIM + gz) * YDIM + gy) * WRDX + wx;
        Eout[gw] = e;
        Dout[gw] = d;

        const uint32_t nd = ~d;
        uint4* ci = (uint4*)(cntIn + (size_t)gw * 8);
        uint4* co = (uint4*)(cntOut + (size_t)gw * 8);
        uint4 a0 = ci[0], a1 = ci[1];
        a0.x += spread4(e & 15u);         a0.y += spread4((e >> 4) & 15u);
        a0.z += spread4((e >> 8) & 15u);  a0.w += spread4((e >> 12) & 15u);
        a1.x += spread4((e >> 16) & 15u); a1.y += spread4((e >> 20) & 15u);
        a1.z += spread4((e >> 24) & 15u); a1.w += spread4((e >> 28) & 15u);
        ci[0] = a0; ci[1] = a1;
        uint4 c0 = co[0], c1 = co[1];
        c0.x += spread4(nd & 15u);         c0.y += spread4((nd >> 4) & 15u);
        c0.z += spread4((nd >> 8) & 15u);  c0.w += spread4((nd >> 12) & 15u);
        c1.x += spread4((nd >> 16) & 15u); c1.y += spread4((nd >> 20) & 15u);
        c1.z += spread4((nd >> 24) & 15u); c1.w += spread4((nd >> 28) & 15u);
        co[0] = c0; co[1] = c1;
    }
}

// ---------------------------------------------------------------- loss
// Streams inputs once (dominant HBM pass). Unrolled batch loop keeps all
// addressing in 32-bit (chunks never cross a batch: VOL % 64 == 0).
// Each wave handles 64-element chunks; one V_WMMA_F32_16X16X4_F32 per chunk
// accumulates p_fg * dist on the C-matrix diagonal.
// A layout (16x4 f32): lane l holds M = l&15; VGPR j holds K = 2*(l>>4)+j.
// B layout (4x16 f32): mirrored, so lane l loads elements
// chunk*64 + (l&15)*4 + 2*(l>>4) + {0,1} for both A and B, and
// C[m][m] += sum_k p[4m+k] * d[4m+k].
// dist select is done arithmetically (d = t*(1-ci-co)+co) so the byte loads
// remain unconditional -> no divergent exec-masked load branches.
__global__ __launch_bounds__(256) void bl_loss(const float* __restrict__ inp,
                                               const uint32_t* __restrict__ mask0,
                                               const uint8_t* __restrict__ ci,
                                               const uint8_t* __restrict__ co,
                                               const int* __restrict__ fg,
                                               float* __restrict__ out) {
    const int lane = threadIdx.x & 31;
    const int wid = blockIdx.x * (blockDim.x >> 5) + (threadIdx.x >> 5);
    const int nw = gridDim.x * (blockDim.x >> 5);
    const int loff = (lane & 15) * 4 + (lane >> 4) * 2;  // even, 0..62
    const int chunksPerB = VOL / 64;                     // 55,296 ; 55,296 % nw == 0

    v8f acc = {};
#pragma unroll
    for (int b = 0; b < BDIM; ++b) {
        const float* ch0 = inp + (size_t)(b * 2) * VOL;
        const float* ch1 = ch0 + VOL;
        const bool nofg = (fg[b] == 0);  // uniform per batch
        const int bv = b * VOL;
        for (int chunk = wid; chunk < chunksPerB; chunk += nw) {
            const int off = chunk * 64 + loff;   // within-batch voxel (even)
            const int eg = bv + off;             // global voxel (even)
            const float2 xa = *(const float2*)(ch0 + off);
            const float2 xb = *(const float2*)(ch1 + off);
            // softmax channel 1 via fast reciprocal
            const float p0 = __builtin_amdgcn_rcpf(1.0f + __expf(xa.x - xb.x));
            const float p1 = __builtin_amdgcn_rcpf(1.0f + __expf(xa.y - xb.y));
            const uint32_t tb = mask0[(uint32_t)eg >> 5] >> (eg & 31);  // bits 0,1
            const uchar2 cin = *(const uchar2*)(ci + (size_t)eg);
            const uchar2 cot = *(const uchar2*)(co + (size_t)eg);
            const float t0 = (float)(tb & 1u);
            const float t1 = (float)((tb >> 1) & 1u);
            const float ci0 = (float)cin.x, ci1 = (float)cin.y;
            const float co0 = (float)cot.x, co1 = (float)cot.y;
            // d = t ? (1 - ci) : co  ==  co + t*(1 - ci - co); both loads used
            float d0 = fmaf(t0, 1.0f - ci0 - co0, co0);
            float d1 = fmaf(t1, 1.0f - ci1 - co1, co1);
            d0 = nofg ? 1.0f : d0;
            d1 = nofg ? 1.0f : d1;
            const v2f A = {p0, p1};
            const v2f B = {d0, d1};
            acc = __builtin_amdgcn_wmma_f32_16x16x4_f32(false, A, false, B,
                                                        (short)0, acc, false, false);
        }
    }
    // diagonal of C: lanes 0..7 -> acc[lane], lanes 24..31 -> acc[lane-24]
    float diag = 0.0f;
#pragma unroll
    for (int r = 0; r < 8; ++r)
        diag += ((lane == r) || (lane == r + 24)) ? acc[r] : 0.0f;
#pragma unroll
    for (int off = 16; off; off >>= 1) diag += __shfl_xor(diag, off, 32);
    if (lane == 0) atomicAdd(out, diag * (1.0f / (float)TOTV));
}

// ---------------------------------------------------------------- launch
extern "C" void kernel_launch(void* const* d_in, const int* in_sizes, int n_in,
                              void* d_out, int out_size, void* d_ws, size_t ws_size,
                              hipStream_t stream) {
    const float* inp = (const float*)d_in[0];   // (4,2,96,192,192) f32
    const int* tgt = (const int*)d_in[1];       // (4,1,96,192,192) i32
    float* out = (float*)d_out;                 // scalar

    char* ws = (char*)d_ws;
    const size_t WB = (size_t)NWORDS * 4;       // 1,769,472 B per mask buffer
    uint32_t* mask0 = (uint32_t*)(ws);
    uint32_t* Ebuf[2] = {(uint32_t*)(ws + WB), (uint32_t*)(ws + 2 * WB)};
    uint32_t* Dbuf[2] = {(uint32_t*)(ws + 3 * WB), (uint32_t*)(ws + 4 * WB)};
    uint32_t* cntIn = (uint32_t*)(ws + 5 * WB);                     // 14,155,776 B (u8/voxel)
    uint32_t* cntOut = (uint32_t*)(ws + 5 * WB + (size_t)TOTV);     // 14,155,776 B
    int* fg = (int*)(ws + 5 * WB + 2 * (size_t)TOTV);               // 4 ints

    bl_init<<<1, 32, 0, stream>>>(fg, out);
    bl_pack<<<NWORDS / 256, 256, 0, stream>>>(tgt, mask0, Ebuf[0], Dbuf[0],
                                              cntIn, cntOut, fg);
    int cur = 0;
    const int eroBlocks = (YDIM / 16) * (ZDIM / 8) * BDIM;  // 576
    for (int it = 0; it < ITERS; ++it) {
        bl_erode<<<eroBlocks, 256, 0, stream>>>(Ebuf[cur], Dbuf[cur],
                                                Ebuf[cur ^ 1], Dbuf[cur ^ 1],
                                                cntIn, cntOut);
        cur ^= 1;
    }
    // 864 blocks * 8 waves = 6912 waves; 55,296 chunks/batch / 6912 = 8 exactly
    bl_loss<<<864, 256, 0, stream>>>(inp, mask0, (const uint8_t*)cntIn,
                                     (const uint8_t*)cntOut, fg, out);
}